// custom_v2_76046690943483
// MI455X (gfx1250) — compile-verified
//
#include <hip/hip_runtime.h>

// Problem constants (match reference)
#define B_     64
#define T_     64
#define S_     128
#define P_     (S_ * S_)   // 16384 pixels
#define U2_    16
#define RECON_ 512

// Tiling
#define PPB    16          // pixels per block, one per wave (16 waves = 512 thr)
#define KC     8           // t-values staged per LDS chunk
#define PITCH  20          // row pitch (floats): 16B-aligned rows + conflict-free
                           // (20*b mod 64 distinct for b=0..15)

typedef __attribute__((ext_vector_type(2))) float v2f;
typedef __attribute__((ext_vector_type(8))) float v8f;
typedef int async_v4i __attribute__((vector_size(16)));  // exact builtin param type

#if defined(__has_builtin)
#if __has_builtin(__builtin_amdgcn_global_load_async_to_lds_b128) && \
    __has_builtin(__builtin_amdgcn_s_wait_asynccnt)
#define USE_ASYNC_LDS 1
#endif
#endif

#define ASYNC_PTR(p) ((async_v4i*)(uintptr_t)(p))

__global__ __launch_bounds__(512)
void upscale_wmma_f32(const float* __restrict__ yt,       // (B,T,S,S)
                      const float* __restrict__ weights,  // (P,T,U2)
                      const float* __restrict__ biases,   // (P,1,U2)
                      float* __restrict__ out)            // (B,1,512,512)
{
    // LDS layout: Ybuf[(t_local*64 + b)*PITCH + pixel], pixel = 0..15
    __shared__ float Ybuf[KC * 64 * PITCH];  // 40,960 bytes

    const int tid    = threadIdx.x;
    const int lane   = tid & 31;
    const int wave   = tid >> 5;     // 0..15 : pixel slot within block
    const int laneLo = lane & 15;    // N index / M row selector
    const int laneHi = lane >> 4;    // half-wave selector

    const int p_base = blockIdx.x * PPB;
    const int p      = p_base + wave;

    // Accumulators: 4 tiles of 16x16 f32 covering b = 0..63 (M) x u = 0..15 (N)
    v8f acc[4] = {};

#ifndef USE_ASYNC_LDS
    const int ldsPP    = tid & 15;   // pixel lane for staging (coalesced along p)
    const int ldsGroup = tid >> 4;   // 0..31
#endif

    for (int c = 0; c < T_ / KC; ++c) {
        const int t0 = c * KC;

#ifdef USE_ASYNC_LDS
        // ---- Stage yt chunk via async DMA to LDS: 2048 x b128, 4/thread.
        // Lanes 0..3 cover one 64B row, so each 4-lane group is contiguous.
        #pragma unroll
        for (int i = 0; i < 4; ++i) {
            int q       = i * 512 + tid;      // 0..2047
            int tb      = q >> 2;             // row: t_local*64 + b
            int pp4     = q & 3;              // 16B sub-row
            int t_local = tb >> 6;
            int b       = tb & 63;
            __builtin_amdgcn_global_load_async_to_lds_b128(
                ASYNC_PTR(&yt[(size_t)(b * T_ + t0 + t_local) * P_ +
                              p_base + pp4 * 4]),
                ASYNC_PTR(&Ybuf[tb * PITCH + pp4 * 4]),
                /*offset=*/0, /*cpol=*/0);
        }
        __builtin_amdgcn_s_wait_asynccnt(0);
#else
        // ---- Fallback: manual staging through VGPRs ----
        #pragma unroll
        for (int i = 0; i < (KC * 64 * PPB) / 512; ++i) {   // 16 iters
            int tb      = i * 32 + ldsGroup;                // 0..511
            int t_local = tb >> 6;
            int b       = tb & 63;
            Ybuf[tb * PITCH + ldsPP] =
                yt[(size_t)(b * T_ + t0 + t_local) * P_ + p_base + ldsPP];
        }
#endif
        // Prefetch next chunk region (global_prefetch_b8)
        if (c + 1 < T_ / KC) {
            __builtin_prefetch(
                &yt[(size_t)((tid >> 4) * T_ + t0 + KC) * P_ + p_base + (tid & 15)],
                0, 1);
        }
        __syncthreads();

        // ---- 2 K-steps of 4 per chunk, 4 M-tiles each ----
        #pragma unroll
        for (int kk = 0; kk < KC / 4; ++kk) {
            const int k0 = t0 + kk * 4;

            // B fragment (4x16): per-pixel weights direct from global.
            // VGPR j holds K = k0 + laneHi*2 + j, N = laneLo.
            v2f bfrag;
            bfrag.x = weights[(size_t)(p * T_ + k0 + laneHi * 2 + 0) * U2_ + laneLo];
            bfrag.y = weights[(size_t)(p * T_ + k0 + laneHi * 2 + 1) * U2_ + laneLo];

            #pragma unroll
            for (int m = 0; m < 4; ++m) {
                // A fragment (16x4): M = m*16 + laneLo, VGPR j -> K = laneHi*2 + j
                const int bRow = m * 16 + laneLo;
                const int tA   = kk * 4 + laneHi * 2;
                v2f afrag;
                afrag.x = Ybuf[((tA + 0) * 64 + bRow) * PITCH + wave];
                afrag.y = Ybuf[((tA + 1) * 64 + bRow) * PITCH + wave];

                acc[m] = __builtin_amdgcn_wmma_f32_16x16x4_f32(
                    /*neg_a=*/false, afrag,
                    /*neg_b=*/false, bfrag,
                    /*c_mod=*/(short)0, acc[m],
                    /*reuse_a=*/false, /*reuse_b=*/false);
            }
        }
        __syncthreads();
    }

    // ---- Epilogue: bias + pixel-shuffle store ----
    // C/D layout: VGPR i -> M = i (lanes 0-15) or i+8 (lanes 16-31); N = laneLo.
    const float biasv = biases[(size_t)p * U2_ + laneLo];

    const int s0 = p >> 7;          // p / S
    const int s1 = p & (S_ - 1);    // p % S
    const int u0 = laneLo >> 2;
    const int u1 = laneLo & 3;
    const size_t rowbase = (size_t)(s0 * 4 + u0) * RECON_ + (s1 * 4 + u1);

    #pragma unroll
    for (int m = 0; m < 4; ++m) {
        #pragma unroll
        for (int i = 0; i < 8; ++i) {
            int b = m * 16 + i + laneHi * 8;
            out[(size_t)b * (RECON_ * RECON_) + rowbase] = acc[m][i] + biasv;
        }
    }
}

extern "C" void kernel_launch(void* const* d_in, const int* in_sizes, int n_in,
                              void* d_out, int out_size, void* d_ws, size_t ws_size,
                              hipStream_t stream) {
    const float* yt      = (const float*)d_in[0];
    const float* weights = (const float*)d_in[1];
    const float* biases  = (const float*)d_in[2];
    float* out           = (float*)d_out;

    dim3 grid(P_ / PPB);   // 1024 blocks
    dim3 block(512);       // 16 waves, one pixel each
    hipLaunchKernelGGL(upscale_wmma_f32, grid, block, 0, stream,
                       yt, weights, biases, out);
}